// KNNSearch_49581102465311
// MI455X (gfx1250) — compile-verified
//
#include <hip/hip_runtime.h>

#define NPTS 32768
#define NQRY 8192
#define KNN  16
#define WPB  8                        // waves per block (point-range split)
#define PTS_PER_WAVE  (NPTS / WPB)    // 4096
#define TILES_PER_WAVE (PTS_PER_WAVE / 16) // 256

typedef float v2f __attribute__((ext_vector_type(2)));
typedef float v8f __attribute__((ext_vector_type(8)));

// ---------------------------------------------------------------------------
// Pack points as [px, py, pz, |p|^2] and queries as [-2qx, -2qy, -2qz, 1].
// Then (Aq) . (Bp) = |p|^2 - 2 q.p  -> one V_WMMA_F32_16X16X4_F32 per tile.
// |q|^2 is a per-row constant (order-preserving); added back at the end.
// ---------------------------------------------------------------------------
__global__ void knn_pack_kernel(const float* __restrict__ points,
                                const float* __restrict__ queries,
                                float4* __restrict__ ppk,
                                float4* __restrict__ qpk,
                                float* __restrict__ q2)
{
    int i = blockIdx.x * blockDim.x + threadIdx.x;
    if (i < NPTS) {
        float x = points[3 * i + 0];
        float y = points[3 * i + 1];
        float z = points[3 * i + 2];
        ppk[i] = make_float4(x, y, z, x * x + y * y + z * z);
    }
    if (i < NQRY) {
        float x = queries[3 * i + 0];
        float y = queries[3 * i + 1];
        float z = queries[3 * i + 2];
        qpk[i] = make_float4(-2.0f * x, -2.0f * y, -2.0f * z, 1.0f);
        q2[i] = x * x + y * y + z * z;
    }
}

__global__ void knn_rowsplits_kernel(float* __restrict__ out)
{
    int i = blockIdx.x * blockDim.x + threadIdx.x;
    if (i <= NQRY) out[NQRY * KNN + i] = (float)(i * KNN);
}

// Fully unrolled sorted insert (constant indices only -> stays in VGPRs).
// List ascending; d[KNN-1] is the current worst. Index payload kept as float.
__device__ __forceinline__ void topk_insert(float (&d)[KNN], float (&ix)[KNN],
                                            float c, float ci)
{
    if (c < d[KNN - 1]) {               // common path: one compare + skip
#pragma unroll
        for (int i = KNN - 1; i >= 1; --i) {
            bool  lt = c < d[i - 1];
            float nd = lt ? d[i - 1] : c;
            float ni = lt ? ix[i - 1] : ci;
            bool  up = c < d[i];
            d[i]  = up ? nd : d[i];
            ix[i] = up ? ni : ix[i];
        }
        if (c < d[0]) { d[0] = c; ix[0] = ci; }
    }
}

// ---------------------------------------------------------------------------
// 256 blocks x 256 threads (8 waves). Each block owns 32 queries; each wave
// scans a disjoint 4096-point range with V_WMMA_F32_16X16X4_F32 (2 per tile,
// 32x16 scores), transposes the score tile through LDS, and keeps a per-lane
// top-16. Partial lists are merged in LDS at the end of the block.
// ---------------------------------------------------------------------------
__launch_bounds__(WPB * 32)
__global__ void knn_search_kernel(const float4* __restrict__ ppk,
                                  const float4* __restrict__ qpk,
                                  const float* __restrict__ q2,
                                  float* __restrict__ out)
{
    __shared__ float tiles[WPB][32 * 20];     // per-wave 32x16 score tile (+pad)
    __shared__ float mdist[32][WPB][KNN];     // per-query partial top-16 dists
    __shared__ float midxf[32][WPB][KNN];     // per-query partial top-16 indices

    const int tid  = threadIdx.x;
    const int wave = tid >> 5;
    const int lane = tid & 31;
    const int l15  = lane & 15;
    const int half = lane >> 4;               // selects K pair {0,1} vs {2,3}
    const int qbase = blockIdx.x * 32;

    // A operands: lane half h holds K=2h (vgpr0), K=2h+1 (vgpr1); M = l15.
    // One global_load_b64 per lane: float2 view, element 2*row + half.
    const float2* qp2 = (const float2*)qpk;
    float2 qa0 = qp2[2 * (qbase + l15) + half];
    float2 qa1 = qp2[2 * (qbase + 16 + l15) + half];
    v2f a0 = {qa0.x, qa0.y};
    v2f a1 = {qa1.x, qa1.y};

    const float2* pp2 = (const float2*)ppk;
    const int prange = wave * PTS_PER_WAVE;

    float dist[KNN];
    float idxf[KNN];
#pragma unroll
    for (int j = 0; j < KNN; ++j) { dist[j] = 3.0e38f; idxf[j] = 0.0f; }

    float* mytile = &tiles[wave][0];

    // Software pipeline: prefetch next tile's point pair before selection work.
    float2 pcur = pp2[2 * (prange + l15) + half];
    for (int t = 0; t < TILES_PER_WAVE; ++t) {
        const int nnext = (t + 1 < TILES_PER_WAVE)
                              ? prange + (t + 1) * 16 + l15
                              : prange + l15;            // harmless reload
        float2 pnext = pp2[2 * nnext + half];

        v2f b = {pcur.x, pcur.y};
        v8f zz = {};
        // D = A x B: 16x16 tile of (|p|^2 - 2 q.p)
        v8f c0 = __builtin_amdgcn_wmma_f32_16x16x4_f32(
            false, a0, false, b, (short)0, zz, false, false);
        v8f c1 = __builtin_amdgcn_wmma_f32_16x16x4_f32(
            false, a1, false, b, (short)0, zz, false, false);

        // C layout: vgpr v, lanes 0-15 -> row v, lanes 16-31 -> row v+8; col=l15
#pragma unroll
        for (int v = 0; v < 8; ++v)
            mytile[(half * 8 + v) * 20 + l15] = c0[v];
#pragma unroll
        for (int v = 0; v < 8; ++v)
            mytile[(16 + half * 8 + v) * 20 + l15] = c1[v];

        // LDS ops are in-order within a wave (DScnt); barrier only pins
        // compiler ordering (emits no hardware instruction).
        __builtin_amdgcn_wave_barrier();

        float sc[16];
#pragma unroll
        for (int j = 0; j < 16; ++j) sc[j] = mytile[lane * 20 + j];

        __builtin_amdgcn_wave_barrier();

        const int pbase = prange + t * 16;
#pragma unroll
        for (int j = 0; j < 16; ++j)
            topk_insert(dist, idxf, sc[j], (float)(pbase + j));

        pcur = pnext;
    }

    // Publish partial lists; merge 8 sorted 16-lists per query.
#pragma unroll
    for (int j = 0; j < KNN; ++j) {
        mdist[lane][wave][j] = dist[j];
        midxf[lane][wave][j] = idxf[j];
    }
    __syncthreads();

    if (tid < 32) {
        const int q = qbase + tid;
        float fd[KNN];
        float fi[KNN];
#pragma unroll
        for (int j = 0; j < KNN; ++j) { fd[j] = 3.0e38f; fi[j] = 0.0f; }

        for (int w = 0; w < WPB; ++w)
            for (int j = 0; j < KNN; ++j)
                topk_insert(fd, fi, mdist[tid][w][j], midxf[tid][w][j]);

        // Flat output (reference tuple order, all as float):
        //   [0, Q*K)                      neighbors_index
        //   [Q*K, Q*K + Q + 1)            neighbors_row_splits
        //   [Q*K + Q + 1, 2*Q*K + Q + 1)  neighbors_distance (ascending)
        const float qn = q2[q];
#pragma unroll
        for (int j = 0; j < KNN; ++j) {
            out[q * KNN + j] = fi[j];
            out[NQRY * KNN + (NQRY + 1) + q * KNN + j] = fd[j] + qn;
        }
    }
}

extern "C" void kernel_launch(void* const* d_in, const int* in_sizes, int n_in,
                              void* d_out, int out_size, void* d_ws, size_t ws_size,
                              hipStream_t stream)
{
    const float* points  = (const float*)d_in[0];   // [32768, 3] f32
    const float* queries = (const float*)d_in[1];   // [8192, 3]  f32
    // d_in[2] = k (fixed 16), d_in[3]/d_in[4] = row splits (single batch) — unused.
    (void)in_sizes; (void)n_in; (void)out_size; (void)ws_size;

    float4* ppk = (float4*)d_ws;                                          // 512 KB
    float4* qpk = (float4*)((char*)d_ws + (size_t)NPTS * sizeof(float4)); // 128 KB
    float*  q2  = (float*)((char*)d_ws +
                           (size_t)(NPTS + NQRY) * sizeof(float4));       //  32 KB
    float* out = (float*)d_out;

    knn_pack_kernel<<<(NPTS + 255) / 256, 256, 0, stream>>>(points, queries,
                                                            ppk, qpk, q2);
    knn_rowsplits_kernel<<<(NQRY + 1 + 255) / 256, 256, 0, stream>>>(out);
    knn_search_kernel<<<NQRY / 32, WPB * 32, 0, stream>>>(ppk, qpk, q2, out);
}